// ResidualUncertaintyEncoder_61692910240546
// MI455X (gfx1250) — compile-verified
//
#include <hip/hip_runtime.h>
#include <hip/hip_bf16.h>
#include <math.h>

typedef __attribute__((ext_vector_type(2))) float v2f;
typedef __attribute__((ext_vector_type(8))) float v8f;

#define H_RAW   3072
#define W_RAW   4096
#define CH_H    1536
#define CH_W    2048
#define NB      2
#define BIGF    3.402823466e38f

__device__ __forceinline__ float softplus_f(float x) {
    // numerically stable softplus: max(x,0) + log1p(exp(-|x|))
    return fmaxf(x, 0.0f) + log1pf(expf(-fabsf(x)));
}

// ---------- bitonic sort of a single 2048-float LDS array, 256 threads ----------
__device__ __forceinline__ void sort2048(float* d, int tid) {
    for (int k = 2; k <= 2048; k <<= 1) {
        for (int j = k >> 1; j > 0; j >>= 1) {
            __syncthreads();
            for (int t = tid; t < 2048; t += 256) {
                int p = t ^ j;
                if (p > t) {
                    float x = d[t], y = d[p];
                    bool up = ((t & k) == 0);
                    if (up ? (x > y) : (x < y)) { d[t] = y; d[p] = x; }
                }
            }
        }
    }
    __syncthreads();
}

// ---------- dual bitonic sort: buf[0..2047] and buf[2048..4095] together ----------
__device__ __forceinline__ void sort2x2048(float* buf, int tid) {
    for (int k = 2; k <= 2048; k <<= 1) {
        for (int j = k >> 1; j > 0; j >>= 1) {
            __syncthreads();
            for (int t = tid; t < 4096; t += 256) {
                int li   = t & 2047;
                int base = t & 2048;
                int p = li ^ j;
                if (p > li) {
                    float x = buf[base + li], y = buf[base + p];
                    bool up = ((li & k) == 0);
                    if (up ? (x > y) : (x < y)) { buf[base + li] = y; buf[base + p] = x; }
                }
            }
        }
    }
    __syncthreads();
}

// =====================================================================
// Pass 1: per-raw-row — deinterleave RGGB pair, row medians + stat partials
// grid = NB*H_RAW blocks (one raw row each), 256 threads
// =====================================================================
__global__ __launch_bounds__(256) void row_stats_kernel(
    const float* __restrict__ raw,
    float* __restrict__ partials,   // [NB*H_RAW][8]: {sAbsE,sSqE,maxE,sAbsO,sSqO,maxO,-,-}
    float* __restrict__ rowmed)     // [NB][4][CH_H]
{
    __shared__ float buf[4096];     // even cols | odd cols
    __shared__ float wred[8][6];
    const int tid  = threadIdx.x;
    const int blk  = blockIdx.x;            // b*H_RAW + row
    const int b    = blk / H_RAW;
    const int row  = blk - b * H_RAW;
    const int half = row & 1;               // 0 -> ch 0,1 ; 1 -> ch 2,3
    const int i    = row >> 1;

    const float4* rp = (const float4*)(raw + (size_t)blk * W_RAW);
    float sa = 0.f, sq = 0.f, mA = 0.f, sb = 0.f, sqb = 0.f, mB = 0.f;
#pragma unroll
    for (int v = 0; v < 4; ++v) {
        int q = tid + v * 256;              // float4 index within row (0..1023)
        __builtin_prefetch(rp + q + 256, 0, 0);
        float4 f = rp[q];
        buf[2 * q]          = f.x;
        buf[2 * q + 1]      = f.z;
        buf[2048 + 2 * q]     = f.y;
        buf[2048 + 2 * q + 1] = f.w;
        float ax = fabsf(f.x), az = fabsf(f.z), ay = fabsf(f.y), aw = fabsf(f.w);
        sa  += ax + az;  sq  += f.x * f.x + f.z * f.z;  mA = fmaxf(mA, fmaxf(ax, az));
        sb  += ay + aw;  sqb += f.y * f.y + f.w * f.w;  mB = fmaxf(mB, fmaxf(ay, aw));
    }
    // wave32 shuffle reduction
    for (int off = 16; off > 0; off >>= 1) {
        sa  += __shfl_down(sa, off);
        sq  += __shfl_down(sq, off);
        sb  += __shfl_down(sb, off);
        sqb += __shfl_down(sqb, off);
        mA = fmaxf(mA, __shfl_down(mA, off));
        mB = fmaxf(mB, __shfl_down(mB, off));
    }
    const int lane = tid & 31, wave = tid >> 5;
    if (lane == 0) {
        wred[wave][0] = sa;  wred[wave][1] = sq;  wred[wave][2] = mA;
        wred[wave][3] = sb;  wred[wave][4] = sqb; wred[wave][5] = mB;
    }
    __syncthreads();
    if (tid == 0) {
        float t0 = 0, t1 = 0, t2 = 0, t3 = 0, t4 = 0, t5 = 0;
        for (int w = 0; w < 8; ++w) {
            t0 += wred[w][0]; t1 += wred[w][1]; t2 = fmaxf(t2, wred[w][2]);
            t3 += wred[w][3]; t4 += wred[w][4]; t5 = fmaxf(t5, wred[w][5]);
        }
        float* p = partials + (size_t)blk * 8;
        p[0] = t0; p[1] = t1; p[2] = t2; p[3] = t3; p[4] = t4; p[5] = t5;
    }

    sort2x2048(buf, tid);
    if (tid == 0) {
        int cA = half * 2, cB = half * 2 + 1;
        rowmed[((size_t)b * 4 + cA) * CH_H + i] = buf[1023];          // lower median of 2048
        rowmed[((size_t)b * 4 + cB) * CH_H + i] = buf[2048 + 1023];
    }
}

// =====================================================================
// Pass 2: per (b,c) — median of row medians + finalize mean/max feats
// grid = 8 blocks, 256 threads
// =====================================================================
__global__ __launch_bounds__(256) void col_med_kernel(
    const float* __restrict__ partials,
    const float* __restrict__ rowmed,
    float* __restrict__ med,        // [NB*4]
    float* __restrict__ feats)      // [NB][16]
{
    __shared__ float buf[2048];
    const int tid = threadIdx.x;
    const int c = blockIdx.x & 3, b = blockIdx.x >> 2;

    float s0 = 0.f, s1 = 0.f, s2 = 0.f;
    for (int i = tid; i < CH_H; i += 256) {
        const float* p = partials + ((size_t)(b * H_RAW + 2 * i + (c >> 1))) * 8 + (c & 1) * 3;
        s0 += p[0]; s1 += p[1]; s2 = fmaxf(s2, p[2]);
    }
    float sumAbs, sumSq, mx;
    buf[tid] = s0; __syncthreads();
    for (int s = 128; s > 0; s >>= 1) { if (tid < s) buf[tid] += buf[tid + s]; __syncthreads(); }
    sumAbs = buf[0]; __syncthreads();
    buf[tid] = s1; __syncthreads();
    for (int s = 128; s > 0; s >>= 1) { if (tid < s) buf[tid] += buf[tid + s]; __syncthreads(); }
    sumSq = buf[0]; __syncthreads();
    buf[tid] = s2; __syncthreads();
    for (int s = 128; s > 0; s >>= 1) { if (tid < s) buf[tid] = fmaxf(buf[tid], buf[tid + s]); __syncthreads(); }
    mx = buf[0]; __syncthreads();

    for (int i = tid; i < 2048; i += 256)
        buf[i] = (i < CH_H) ? rowmed[((size_t)b * 4 + c) * CH_H + i] : BIGF;
    sort2048(buf, tid);
    if (tid == 0) {
        med[b * 4 + c] = buf[(CH_H - 1) / 2];        // 767: lower median of 1536
        const float invN = 1.0f / ((float)CH_H * (float)CH_W);
        feats[b * 16 + c]       = sumAbs * invN;     // mean_abs
        feats[b * 16 + 4 + c]   = sumSq * invN;      // mean_r2
        feats[b * 16 + 12 + c]  = mx;                // max_abs
    }
}

// =====================================================================
// Pass 3: per-raw-row — row medians of |x - med[b][c]|
// =====================================================================
__global__ __launch_bounds__(256) void row_mad_kernel(
    const float* __restrict__ raw,
    const float* __restrict__ med,
    float* __restrict__ rowmad)     // [NB][4][CH_H]
{
    __shared__ float buf[4096];
    const int tid  = threadIdx.x;
    const int blk  = blockIdx.x;
    const int b    = blk / H_RAW;
    const int row  = blk - b * H_RAW;
    const int half = row & 1;
    const int i    = row >> 1;
    const float medE = med[b * 4 + half * 2];
    const float medO = med[b * 4 + half * 2 + 1];

    const float4* rp = (const float4*)(raw + (size_t)blk * W_RAW);
#pragma unroll
    for (int v = 0; v < 4; ++v) {
        int q = tid + v * 256;
        __builtin_prefetch(rp + q + 256, 0, 0);
        float4 f = rp[q];
        buf[2 * q]            = fabsf(f.x - medE);
        buf[2 * q + 1]        = fabsf(f.z - medE);
        buf[2048 + 2 * q]     = fabsf(f.y - medO);
        buf[2048 + 2 * q + 1] = fabsf(f.w - medO);
    }
    sort2x2048(buf, tid);
    if (tid == 0) {
        int cA = half * 2, cB = half * 2 + 1;
        rowmad[((size_t)b * 4 + cA) * CH_H + i] = buf[1023];
        rowmad[((size_t)b * 4 + cB) * CH_H + i] = buf[2048 + 1023];
    }
}

// =====================================================================
// Pass 4: per (b,c) — median of row MADs -> mad_vars feat
// =====================================================================
__global__ __launch_bounds__(256) void col_mad_kernel(
    const float* __restrict__ rowmad,
    float* __restrict__ feats)
{
    __shared__ float buf[2048];
    const int tid = threadIdx.x;
    const int c = blockIdx.x & 3, b = blockIdx.x >> 2;
    for (int i = tid; i < 2048; i += 256)
        buf[i] = (i < CH_H) ? rowmad[((size_t)b * 4 + c) * CH_H + i] : BIGF;
    sort2048(buf, tid);
    if (tid == 0) {
        float mad = buf[(CH_H - 1) / 2];
        float sg = 1.4826f * mad;
        feats[b * 16 + 8 + c] = sg * sg + 1e-8f;     // mad_vars
    }
}

// =====================================================================
// MLP: feats(2,16 padded to 16x16) through both softplus MLPs with
// V_WMMA_F32_16X16X4_F32.  1 block x 256 threads (8 waves).
// =====================================================================
__global__ __launch_bounds__(256) void mlp_kernel(
    const float* __restrict__ feats,
    const float* __restrict__ eW1, const float* __restrict__ eb1,
    const float* __restrict__ eW2, const float* __restrict__ eb2,
    const float* __restrict__ uW1, const float* __restrict__ ub1,
    const float* __restrict__ uW2, const float* __restrict__ ub2,
    const float* __restrict__ gainp,
    float* __restrict__ out)        // [2*64 emb][2 u]
{
    __shared__ float Hs[2 * 16 * 128];        // H_e | H_u  (16 KB)
    __shared__ float Bs[128 * 64 + 128];      // phase1: B1[2][16][128]; phase3: B2e[128][64] + B2u[128]
    const int tid  = threadIdx.x;
    const int wave = tid >> 5;
    const int lane = tid & 31;
    const int hi   = lane >> 4;               // K-half selector (A/B), M-half (C/D)
    const int ln   = lane & 15;               // M for A, N for B/C/D

    // ---- phase 1: B1[s][k][n] = softplus(W1_s[n][k]) ----
    for (int idx = tid; idx < 2 * 16 * 128; idx += 256) {
        int s = idx >> 11;
        int rem = idx & 2047;
        int k = rem >> 7;
        int n = rem & 127;
        const float* W = s ? uW1 : eW1;
        Bs[idx] = softplus_f(W[n * 16 + k]);
    }
    __syncthreads();

    // ---- phase 2: GEMM1 (16x16 A = padded feats) x (16x128) for s=0(emb),1(u) ----
    for (int s = 0; s < 2; ++s) {
        v8f acc = {0.f, 0.f, 0.f, 0.f, 0.f, 0.f, 0.f, 0.f};
        const float* B1 = Bs + s * 2048;
#pragma unroll
        for (int k0 = 0; k0 < 16; k0 += 4) {
            int ka = k0 + 2 * hi;
            v2f a, bm;
            a.x = (ln < NB) ? feats[ln * 16 + ka]     : 0.f;
            a.y = (ln < NB) ? feats[ln * 16 + ka + 1] : 0.f;
            bm.x = B1[ka * 128 + wave * 16 + ln];
            bm.y = B1[(ka + 1) * 128 + wave * 16 + ln];
            acc = __builtin_amdgcn_wmma_f32_16x16x4_f32(
                false, a, false, bm, (short)0, acc, false, false);
        }
        const float* b1 = s ? ub1 : eb1;
        int col = wave * 16 + ln;
        float bias = b1[col];
#pragma unroll
        for (int r = 0; r < 8; ++r) {
            int m = r + 8 * hi;
            Hs[s * 2048 + m * 128 + col] = softplus_f(acc[r] + bias);
        }
    }
    __syncthreads();

    // ---- phase 3: B2e[k][n] = softplus(eW2[n][k]); B2u[k] = softplus(uW2[k]) ----
    for (int idx = tid; idx < 128 * 64; idx += 256) {
        int k = idx >> 6, n = idx & 63;
        Bs[idx] = softplus_f(eW2[n * 128 + k]);
    }
    if (tid < 128) Bs[128 * 64 + tid] = softplus_f(uW2[tid]);
    __syncthreads();

    // ---- phase 4: GEMM2 ----
    if (wave < 4) {                 // emb: (16x128) x (128x64), tile = wave
        v8f acc = {0.f, 0.f, 0.f, 0.f, 0.f, 0.f, 0.f, 0.f};
        for (int k0 = 0; k0 < 128; k0 += 4) {
            int ka = k0 + 2 * hi;
            v2f a, bm;
            a.x = Hs[ln * 128 + ka];
            a.y = Hs[ln * 128 + ka + 1];
            bm.x = Bs[ka * 64 + wave * 16 + ln];
            bm.y = Bs[(ka + 1) * 64 + wave * 16 + ln];
            acc = __builtin_amdgcn_wmma_f32_16x16x4_f32(
                false, a, false, bm, (short)0, acc, false, false);
        }
        if (hi == 0) {              // VGPR r, lanes 0-15 -> row M=r
            float g = gainp[0];
            int n = wave * 16 + ln;
            out[0 * 64 + n] = g * (acc[0] + eb2[n]);
            out[1 * 64 + n] = g * (acc[1] + eb2[n]);
        }
    } else if (wave == 4) {         // u: (16x128) x (128x16 padded, col 0 real)
        v8f acc = {0.f, 0.f, 0.f, 0.f, 0.f, 0.f, 0.f, 0.f};
        for (int k0 = 0; k0 < 128; k0 += 4) {
            int ka = k0 + 2 * hi;
            v2f a, bm;
            a.x = Hs[2048 + ln * 128 + ka];
            a.y = Hs[2048 + ln * 128 + ka + 1];
            bm.x = (ln == 0) ? Bs[128 * 64 + ka]     : 0.f;
            bm.y = (ln == 0) ? Bs[128 * 64 + ka + 1] : 0.f;
            acc = __builtin_amdgcn_wmma_f32_16x16x4_f32(
                false, a, false, bm, (short)0, acc, false, false);
        }
        if (lane == 0) {
            out[128] = acc[0] + ub2[0];
            out[129] = acc[1] + ub2[0];
        }
    }
}

extern "C" void kernel_launch(void* const* d_in, const int* in_sizes, int n_in,
                              void* d_out, int out_size, void* d_ws, size_t ws_size,
                              hipStream_t stream) {
    (void)in_sizes; (void)n_in; (void)out_size; (void)ws_size;
    const float* r_raw = (const float*)d_in[0];
    const float* eW1   = (const float*)d_in[1];
    const float* eb1   = (const float*)d_in[2];
    const float* eW2   = (const float*)d_in[3];
    const float* eb2   = (const float*)d_in[4];
    const float* uW1   = (const float*)d_in[5];
    const float* ub1   = (const float*)d_in[6];
    const float* uW2   = (const float*)d_in[7];
    const float* ub2   = (const float*)d_in[8];
    const float* gain  = (const float*)d_in[9];
    float* out = (float*)d_out;

    char* ws = (char*)d_ws;
    float* med      = (float*)(ws);                      // 8 floats
    float* feats    = (float*)(ws + 64);                 // 32 floats
    float* partials = (float*)(ws + 256);                // NB*H_RAW*8 floats (192 KB)
    float* rowmed   = (float*)(ws + 256 + (size_t)NB * H_RAW * 8 * 4);  // 48 KB
    float* rowmad   = rowmed;   // reused after col_med consumed rowmed

    row_stats_kernel<<<NB * H_RAW, 256, 0, stream>>>(r_raw, partials, rowmed);
    col_med_kernel  <<<NB * 4,     256, 0, stream>>>(partials, rowmed, med, feats);
    row_mad_kernel  <<<NB * H_RAW, 256, 0, stream>>>(r_raw, med, rowmad);
    col_mad_kernel  <<<NB * 4,     256, 0, stream>>>(rowmad, feats);
    mlp_kernel<<<1, 256, 0, stream>>>(feats, eW1, eb1, eW2, eb2,
                                      uW1, ub1, uW2, ub2, gain, out);
}